// GaussianVectorQuantizer_58420145160552
// MI455X (gfx1250) — compile-verified
//
#include <hip/hip_runtime.h>
#include <hip/hip_bf16.h>
#include <math.h>

// ---------------------------------------------------------------------------
// GaussianVectorQuantizer (eval path), MI455X / gfx1250
//
// Shapes: b=128, npts=1024, dim=64, n_clusters=16, book_size=1024
// Outputs (flat concat): zq[128*1024*64], precision_q[1],
//                        logits[128*1024*1024], mu_sampled[128*1024*64]
// ---------------------------------------------------------------------------

typedef __attribute__((ext_vector_type(2))) float v2f;
typedef __attribute__((ext_vector_type(8))) float v8f;

#define NROWS     131072          // b * npts
#define DIM       64
#define BOOK      1024
#define NCLUST    16

#define ZQ_OFF    ((size_t)0)
#define PREC_OFF  ((size_t)8388608)          // 128*1024*64
#define LOG_OFF   ((size_t)8388609)
#define MU_OFF    ((size_t)142606337)        // 8388609 + 128*1024*1024

#define EPS_CLAMP 1e-10f

// --------------------------------------------------------------------------
// Kernel 1: per-batch argmax of c_probs -> c_idx ; also precision_q output.
// --------------------------------------------------------------------------
__global__ void gvq_prep_scalar(const float* __restrict__ c_probs,
                                const float* __restrict__ log_param_q,
                                int* __restrict__ c_idx,
                                float* __restrict__ out_prec) {
  const int b = threadIdx.x;
  if (b < 128) {
    const float* p = c_probs + b * NCLUST;
    float bv = p[0];
    int bi = 0;
#pragma unroll
    for (int j = 1; j < NCLUST; ++j) {
      const float v = p[j];
      if (v > bv) { bv = v; bi = j; }
    }
    c_idx[b] = bi;
  }
  if (b == 0) {
    out_prec[0] = 0.5f / fmaxf(__expf(log_param_q[0]), EPS_CLAMP);
  }
}

// --------------------------------------------------------------------------
// Kernel 2: transpose book -> bookT [64][1024] (for coalesced B fragments)
//           and column norms ||book_j||^2.
// --------------------------------------------------------------------------
__global__ void gvq_prep_book(const float* __restrict__ book,
                              float* __restrict__ bookT,
                              float* __restrict__ bnorm) {
  const int j = blockIdx.x * blockDim.x + threadIdx.x;
  if (j >= BOOK) return;
  float s = 0.f;
#pragma unroll 8
  for (int k = 0; k < DIM; ++k) {
    const float v = book[(size_t)j * DIM + k];
    s += v * v;
    bookT[(size_t)k * BOOK + j] = v;
  }
  bnorm[j] = s;
}

// --------------------------------------------------------------------------
// Kernel 3: zsum = z + mu[c_idx]  (written into the zq output region, later
// overwritten by the GEMM kernel), and mu_sampled output.
// --------------------------------------------------------------------------
__global__ void gvq_prep_z(const float* __restrict__ z,
                           const float* __restrict__ mu,
                           const int* __restrict__ c_idx,
                           float* __restrict__ zsum,     // = out + ZQ_OFF (16B aligned)
                           float* __restrict__ mu_out) { // = out + MU_OFF (4B aligned)
  const size_t r = (size_t)blockIdx.x * blockDim.x + threadIdx.x;
  const int b  = (int)(r >> 10);
  const int pt = (int)(r & 1023);
  const int ci = c_idx[b];
  const float* zr = z + r * DIM;
  const float* mr = mu + ((size_t)ci * 1024 + pt) * DIM;
  float* so = zsum + r * DIM;
  float* mo = mu_out + r * DIM;
#pragma unroll
  for (int d = 0; d < DIM; d += 4) {
    const float4 zv = *(const float4*)(zr + d);
    const float4 mv = *(const float4*)(mr + d);
    float4 s;
    s.x = zv.x + mv.x; s.y = zv.y + mv.y; s.z = zv.z + mv.z; s.w = zv.w + mv.w;
    *(float4*)(so + d) = s;
    // mu_out base is only 4B aligned (odd float offset) -> scalar stores
    mo[d + 0] = mv.x; mo[d + 1] = mv.y; mo[d + 2] = mv.z; mo[d + 3] = mv.w;
  }
}

// --------------------------------------------------------------------------
// Kernel 4: fused WMMA GEMM + logits + row-argmax + codebook gather.
// One workgroup (8 waves, 256 threads) owns 16 rows x 1024 columns, K=64.
// --------------------------------------------------------------------------
__global__ void __launch_bounds__(256)
gvq_gemm(const float* __restrict__ zsum,   // rows staged from zq region
         const float* __restrict__ bookT,  // [64][1024]
         const float* __restrict__ bnorm,  // [1024]
         const float* __restrict__ book,   // [1024][64]
         const float* __restrict__ log_param_q,
         float* __restrict__ out) {
  const int m0   = blockIdx.x * 16;
  const int t    = threadIdx.x;
  const int wv   = t >> 5;        // wave 0..7
  const int lane = t & 31;
  const int l16  = lane & 15;
  const int hi   = lane >> 4;     // 0 or 1

  __shared__ float a_tile[16 * 68];   // padded stride 68 -> conflict free
  __shared__ float s_znorm[16];
  __shared__ float s_bmax[16 * 128];
  __shared__ int   s_bidx[16 * 128];
  __shared__ int   s_sel[16];

  // Stage the 16x64 A tile (float4, 4 elements per thread).
  {
    const int row = t >> 4;          // 0..15
    const int c4  = (t & 15) << 2;   // 0..60
    const float4 v = *(const float4*)(zsum + (size_t)(m0 + row) * DIM + c4);
    *(float4*)(&a_tile[row * 68 + c4]) = v;
  }
  __syncthreads();

  // Row norms from the staged tile.
  if (t < 16) {
    float s = 0.f;
#pragma unroll
    for (int k = 0; k < DIM; ++k) {
      const float x = a_tile[t * 68 + k];
      s += x * x;
    }
    s_znorm[t] = s;
  }
  __syncthreads();

  const float prec = 0.5f / fmaxf(__expf(log_param_q[0]), EPS_CLAMP);
  float* __restrict__ logits = out + LOG_OFF;

  float best[8];
  int   bidx[8];
#pragma unroll
  for (int p = 0; p < 8; ++p) { best[p] = -3.4e38f; bidx[p] = 0; }

  // A-fragment base for V_WMMA_F32_16X16X4_F32:
  //   VGPR0: K=k0   (lanes 0-15) / K=k0+2 (lanes 16-31)
  //   VGPR1: K=k0+1 (lanes 0-15) / K=k0+3 (lanes 16-31)
  const int arow = l16 * 68 + 2 * hi;

  // Each wave handles column tiles wv, wv+8, ..., wv+56 (uniform -> EXEC all 1s).
  for (int ct = wv; ct < BOOK / 16; ct += 8) {
    const int n0 = ct * 16;
    const float bn = bnorm[n0 + l16];

    v8f acc = {0.f, 0.f, 0.f, 0.f, 0.f, 0.f, 0.f, 0.f};
#pragma unroll
    for (int kk = 0; kk < 16; ++kk) {
      const int k0 = kk * 4;
      v2f a, bf;
      a.x  = a_tile[arow + k0];
      a.y  = a_tile[arow + k0 + 1];
      bf.x = bookT[(size_t)(k0 + 2 * hi)     * BOOK + n0 + l16];
      bf.y = bookT[(size_t)(k0 + 2 * hi + 1) * BOOK + n0 + l16];
      acc = __builtin_amdgcn_wmma_f32_16x16x4_f32(
          /*neg_a=*/false, a, /*neg_b=*/false, bf,
          /*c_mod=*/(short)0, acc, /*reuse_a=*/false, /*reuse_b=*/false);
    }

    // Epilogue: logits = (2*dot - ||z||^2 - ||b||^2) * precision_q
    const int n = n0 + l16;
#pragma unroll
    for (int p = 0; p < 8; ++p) {
      const int m = p + 8 * hi;
      const float val = (2.f * acc[p] - s_znorm[m] - bn) * prec;
      __builtin_nontemporal_store(val, &logits[(size_t)(m0 + m) * BOOK + n]);
      if (val > best[p]) { best[p] = val; bidx[p] = n; }
    }
  }

  // Publish per-lane argmax candidates: 16 rows x (8 waves * 16 lanes) slots.
  {
    const int slot = wv * 16 + l16;
#pragma unroll
    for (int p = 0; p < 8; ++p) {
      const int m = p + 8 * hi;
      s_bmax[m * 128 + slot] = best[p];
      s_bidx[m * 128 + slot] = bidx[p];
    }
  }
  __syncthreads();

  // Reduce 128 candidates per row (ties -> lowest column index, like argmax).
  if (t < 16) {
    float bv = -3.4e38f;
    int bi = BOOK;
    for (int s = 0; s < 128; ++s) {
      const float v = s_bmax[t * 128 + s];
      const int   i = s_bidx[t * 128 + s];
      if (v > bv || (v == bv && i < bi)) { bv = v; bi = i; }
    }
    s_sel[t] = bi;
  }
  __syncthreads();

  // Gather zq = book[argmax] (overwrites this workgroup's own zsum rows).
  {
    const int row = t >> 4;
    const int d0  = (t & 15) << 2;
    const float4 v = *(const float4*)(book + (size_t)s_sel[row] * DIM + d0);
    *(float4*)(out + ZQ_OFF + (size_t)(m0 + row) * DIM + d0) = v;
  }
}

// --------------------------------------------------------------------------
extern "C" void kernel_launch(void* const* d_in, const int* in_sizes, int n_in,
                              void* d_out, int out_size, void* d_ws, size_t ws_size,
                              hipStream_t stream) {
  (void)in_sizes; (void)n_in; (void)out_size; (void)ws_size;

  const float* z       = (const float*)d_in[0];
  const float* c_probs = (const float*)d_in[1];
  const float* mu      = (const float*)d_in[2];
  const float* lpq     = (const float*)d_in[3];
  // d_in[4] = temperature (unused on eval path), d_in[6] = is_train (static 0)
  const float* book    = (const float*)d_in[5];
  float* out = (float*)d_out;

  // Workspace layout (< 300 KB):
  int*   c_idx = (int*)d_ws;                              // 128 ints
  float* bnorm = (float*)((char*)d_ws + 512);             // 1024 floats
  float* bookT = (float*)((char*)d_ws + 8192);            // 64*1024 floats

  gvq_prep_scalar<<<1, 128, 0, stream>>>(c_probs, lpq, c_idx, out + PREC_OFF);
  gvq_prep_book<<<(BOOK + 255) / 256, 256, 0, stream>>>(book, bookT, bnorm);
  gvq_prep_z<<<NROWS / 256, 256, 0, stream>>>(z, mu, c_idx,
                                              out + ZQ_OFF, out + MU_OFF);
  gvq_gemm<<<NROWS / 16, 256, 0, stream>>>(out + ZQ_OFF, bookT, bnorm, book,
                                           lpq, out);
}